// WeightedSampleCrossEntropyLoss_7782480740624
// MI455X (gfx1250) — compile-verified
//
#include <hip/hip_runtime.h>
#include <cstdint>

#define TPB 256
#define NWAVE (TPB / 32)
#define VEC 4                              /* float4 loads per lane per chunk */
#define SEG_FLOATS (TPB * 4)               /* 1024 floats per b128 sweep      */
#define CHUNK_FLOATS (SEG_FLOATS * VEC)    /* 4096 floats = 16 KB per chunk   */

#if defined(__gfx1250__) &&                                                    \
    __has_builtin(__builtin_amdgcn_global_load_async_to_lds_b128) &&           \
    __has_builtin(__builtin_amdgcn_s_wait_asynccnt)
#define USE_ASYNC_LDS 1
#else
#define USE_ASYNC_LDS 0
#endif

#if USE_ASYNC_LDS
typedef int v4i_t __attribute__((vector_size(16)));
// pointer to v4i in global (AS1) / LDS (AS3) address space
typedef __attribute__((address_space(1))) v4i_t* g_v4i_p;
typedef __attribute__((address_space(3))) v4i_t* l_v4i_p;
#endif

// One-exp online softmax update (branch taken only on a new running max,
// which happens O(log n) times per thread -> ~1 v_exp_f32 per element).
__device__ __forceinline__ void fold(float& m, float& s, float x) {
  if (x > m) {
    s = s * __expf(m - x) + 1.0f;
    m = x;
  } else {
    s += __expf(x - m);
  }
}

// Merge two (max, sum) partials.
__device__ __forceinline__ void combine(float& m, float& s, float mo, float so) {
  float mn = fmaxf(m, mo);
  s = s * __expf(m - mn) + so * __expf(mo - mn);
  m = mn;
}

__global__ __launch_bounds__(TPB) void row_loss_kernel(
    const float* __restrict__ logits, const float* __restrict__ weights,
    const int* __restrict__ labels, const int* __restrict__ attrs,
    float* __restrict__ row_loss, int C) {
  const int b = blockIdx.x;
  const int tid = threadIdx.x;
  const float* __restrict__ row = logits + (size_t)b * (size_t)C;

  float m = -3.402823466e38f;
  float s = 0.0f;

  // Rows start 4-byte aligned but not 16-byte aligned (C % 4 == 1).
  // Peel 0..3 head elements so the streamed body is 16-byte aligned.
  const int head =
      (int)(((16u - ((uint32_t)(uintptr_t)row & 15u)) & 15u) >> 2);
  if (tid < head) fold(m, s, row[tid]);

  const float* __restrict__ abase = row + head;
  const int n_al = C - head;
  const int nchunk = n_al / CHUNK_FLOATS;

  // Scalar tail (< CHUNK_FLOATS elements), direct global loads.
  for (int i = head + nchunk * CHUNK_FLOATS + tid; i < C; i += TPB)
    fold(m, s, row[i]);

#if USE_ASYNC_LDS
  // Double-buffered async global->LDS streaming (ASYNCcnt path).
  // Each lane DMAs VEC * 16B per chunk -> 4 async b128 ops per wave per
  // chunk, 2 chunks in flight => up to 4 KB outstanding per wave.
  __shared__ __align__(16) float buf[2][CHUNK_FLOATS];
  auto issue = [&](int c, int slot) {
    const size_t cb = (size_t)c * CHUNK_FLOATS;
#pragma unroll
    for (int j = 0; j < VEC; ++j) {
      const int off = j * SEG_FLOATS + (tid << 2);
      __builtin_amdgcn_global_load_async_to_lds_b128(
          (g_v4i_p)(void*)(float*)(abase + cb + off),
          (l_v4i_p)(void*)&buf[slot][off],
          0, 0);
    }
  };
  if (nchunk > 0) {
    issue(0, 0);
    for (int c = 0; c < nchunk; ++c) {
      if (c + 1 < nchunk) {
        issue(c + 1, (c + 1) & 1);
        __builtin_amdgcn_s_wait_asynccnt(VEC); // chunk c fully landed in LDS
      } else {
        __builtin_amdgcn_s_wait_asynccnt(0);
      }
      // Each lane reads back exactly the bytes it DMA'd: no barrier needed,
      // per-wave ASYNCcnt ordering is sufficient.
#pragma unroll
      for (int j = 0; j < VEC; ++j) {
        const float4 v =
            *(const float4*)&buf[c & 1][j * SEG_FLOATS + (tid << 2)];
        fold(m, s, v.x);
        fold(m, s, v.y);
        fold(m, s, v.z);
        fold(m, s, v.w);
      }
    }
  }
#else
  for (int c = 0; c < nchunk; ++c) {
#pragma unroll
    for (int j = 0; j < VEC; ++j) {
      const float4 v = *(const float4*)(abase + (size_t)c * CHUNK_FLOATS +
                                        j * SEG_FLOATS + (tid << 2));
      fold(m, s, v.x);
      fold(m, s, v.y);
      fold(m, s, v.z);
      fold(m, s, v.w);
    }
  }
#endif

  // wave32 reduction of (m, s)
  for (int off = 16; off > 0; off >>= 1) {
    float mo = __shfl_xor(m, off, 32);
    float so = __shfl_xor(s, off, 32);
    combine(m, s, mo, so);
  }

  __shared__ float lm[NWAVE];
  __shared__ float ls[NWAVE];
  if ((tid & 31) == 0) {
    lm[tid >> 5] = m;
    ls[tid >> 5] = s;
  }
  __syncthreads();

  if (tid == 0) {
    float mt = lm[0], st = ls[0];
#pragma unroll
    for (int i = 1; i < NWAVE; ++i) combine(mt, st, lm[i], ls[i]);
    const int lab = labels[b];
    const int at = attrs[b];
    const float xl = row[lab];
    const float lse = mt + __logf(st);
    row_loss[b] = weights[at * C + lab] * (lse - xl);
  }
}

__global__ __launch_bounds__(TPB) void reduce_mean_kernel(
    const float* __restrict__ rl, float* __restrict__ out, int B) {
  float acc = 0.0f;
  for (int i = threadIdx.x; i < B; i += TPB) acc += rl[i];
  for (int off = 16; off > 0; off >>= 1) acc += __shfl_xor(acc, off, 32);
  __shared__ float lw[NWAVE];
  if ((threadIdx.x & 31) == 0) lw[threadIdx.x >> 5] = acc;
  __syncthreads();
  if (threadIdx.x == 0) {
    float t = 0.0f;
#pragma unroll
    for (int i = 0; i < NWAVE; ++i) t += lw[i];
    out[0] = t / (float)B;
  }
}

extern "C" void kernel_launch(void* const* d_in, const int* in_sizes, int n_in,
                              void* d_out, int out_size, void* d_ws,
                              size_t ws_size, hipStream_t stream) {
  const float* logits = (const float*)d_in[0];   // [B, C] f32
  const float* weights = (const float*)d_in[1];  // [N_ATTR, C] f32
  const int* labels = (const int*)d_in[2];       // [B]
  const int* attrs = (const int*)d_in[3];        // [B]
  const int B = in_sizes[2];
  const int C = in_sizes[0] / B;

  float* row_loss = (float*)d_ws; // B floats of scratch

  row_loss_kernel<<<B, TPB, 0, stream>>>(logits, weights, labels, attrs,
                                         row_loss, C);
  reduce_mean_kernel<<<1, TPB, 0, stream>>>(row_loss, (float*)d_out, B);
}